// PostProcessor_18975165514463
// MI455X (gfx1250) — compile-verified
//
#include <hip/hip_runtime.h>
#include <stdint.h>

#define NCLS        80
#define NANCH       5
#define BB          16
#define HH          76
#define WW          76
#define N_CAND      (BB*HH*WW*NANCH)      /* 462080 */
#define PER_BLOCK   128
#define SCORE_THR   0.02f
#define IOU_THR     0.5f
#define MAXDET      10

// ---------- monotone float <-> ordered-uint mapping (for 64-bit argmax keys) ----------
__device__ __forceinline__ unsigned f2ord(float f) {
    unsigned u = __float_as_uint(f);
    return (u & 0x80000000u) ? ~u : (u | 0x80000000u);
}
__device__ __forceinline__ float ord2f(unsigned o) {
    unsigned u = (o & 0x80000000u) ? (o & 0x7FFFFFFFu) : ~o;
    return __uint_as_float(u);
}

// ---------- CDNA5 async global -> LDS copy (16B per lane) ----------
typedef __attribute__((ext_vector_type(4))) int v4i_t;
typedef __attribute__((address_space(1))) v4i_t* gv4_t;   // global int4*
typedef __attribute__((address_space(3))) v4i_t* lv4_t;   // LDS int4*

__device__ __forceinline__ void async_copy_b128(const float* g, float* l) {
#if __has_builtin(__builtin_amdgcn_global_load_async_to_lds_b128)
    __builtin_amdgcn_global_load_async_to_lds_b128((gv4_t)g, (lv4_t)l, 0, 0);
#else
    unsigned lds_off = (unsigned)(uintptr_t)l;                 // low 32 bits = LDS byte offset
    unsigned long long ga = (unsigned long long)(uintptr_t)g;  // 64-bit global address
    asm volatile("global_load_async_to_lds_b128 %0, %1, off"
                 :: "v"(lds_off), "v"(ga) : "memory");
#endif
}
__device__ __forceinline__ void wait_async0() {
#if __has_builtin(__builtin_amdgcn_s_wait_asynccnt)
    __builtin_amdgcn_s_wait_asynccnt(0);
#else
    asm volatile("s_wait_asynccnt 0" ::: "memory");
#endif
}

__device__ __forceinline__ float sigmoidf_(float x) {
    return 1.0f / (1.0f + __expf(-x));
}

// ================= Kernel 1: decode feats -> (scores_work, boxes, classes) =================
__global__ __launch_bounds__(PER_BLOCK)
void yolo_decode(const float* __restrict__ feats, const float* __restrict__ anchors,
                 float* __restrict__ sw, float4* __restrict__ boxes, float* __restrict__ cls)
{
    __shared__ float lds[PER_BLOCK * 85];   // 43,520 bytes (well under 320KB/WGP)
    const int t = threadIdx.x;
    const int nBase = blockIdx.x * PER_BLOCK;
    const float* gsrc = feats + (size_t)nBase * 85;   // 43,520B-aligned block base

    // Stage 128 anchors * 85 floats = 2720 float4 via async tensor-path loads
    const int NV4 = PER_BLOCK * 85 / 4;               // 2720
    #pragma unroll
    for (int j = 0; j < (NV4 + PER_BLOCK - 1) / PER_BLOCK; ++j) {
        int idx = j * PER_BLOCK + t;
        if (idx < NV4) async_copy_b128(gsrc + idx * 4, &lds[idx * 4]);
    }
    wait_async0();
    __syncthreads();

    // One anchor per thread
    const float* f = &lds[t * 85];
    const int n = nBase + t;
    const int a    = n % NANCH;
    const int wcol = (n / NANCH) % WW;
    const int hrow = (n / (NANCH * WW)) % HH;

    const float aw = anchors[a * 2 + 0];
    const float ah = anchors[a * 2 + 1];

    const float xc = (sigmoidf_(f[0]) + (float)wcol) * (1.0f / (float)WW);
    const float yc = (sigmoidf_(f[1]) + (float)hrow) * (1.0f / (float)HH);
    const float bw = __expf(f[2]) * aw * (1.0f / (float)WW);
    const float bh = __expf(f[3]) * ah * (1.0f / (float)HH);
    const float conf = sigmoidf_(f[4]);

    // argmax over class logits (first occurrence on ties, like jnp.argmax)
    float m = f[5]; int am = 0;
    #pragma unroll 8
    for (int c = 1; c < NCLS; ++c) {
        float v = f[5 + c];
        if (v > m) { m = v; am = c; }
    }
    // max softmax prob = 1 / sum(exp(l - lmax)); score = conf * that
    float se = 0.0f;
    #pragma unroll 8
    for (int c = 0; c < NCLS; ++c) se += __expf(f[5 + c] - m);
    const float score = conf / se;

    const float y1 = yc - 0.5f * bh, x1 = xc - 0.5f * bw;
    const float y2 = yc + 0.5f * bh, x2 = xc + 0.5f * bw;

    boxes[n] = make_float4(y1, x1, y2, x2);
    cls[n]   = (float)am;
    sw[n]    = (score >= SCORE_THR) ? score : -1.0f;
}

// ================= Kernel 2: reset argmax key =================
__global__ void nms_reset(unsigned long long* key) { *key = 0ull; }

// ================= Kernel 3: grid-wide argmax (max score, then min index) =================
__global__ __launch_bounds__(256)
void nms_argmax(const float* __restrict__ sw, unsigned long long* key, int n)
{
    __shared__ unsigned long long red[256];
    unsigned long long best = 0ull;
    for (int i = blockIdx.x * blockDim.x + threadIdx.x; i < n; i += gridDim.x * blockDim.x) {
        unsigned long long k = ((unsigned long long)f2ord(sw[i]) << 32)
                             | (unsigned long long)(0xFFFFFFFFu - (unsigned)i);
        if (k > best) best = k;
    }
    red[threadIdx.x] = best;
    __syncthreads();
    for (int s = 128; s > 0; s >>= 1) {
        if (threadIdx.x < s) {
            if (red[threadIdx.x + s] > red[threadIdx.x]) red[threadIdx.x] = red[threadIdx.x + s];
        }
        __syncthreads();
    }
    if (threadIdx.x == 0) atomicMax(key, red[0]);
}

// ================= Kernel 4: write output row + suppress by IoU =================
__global__ __launch_bounds__(256)
void nms_suppress(float* __restrict__ sw, const float4* __restrict__ boxes,
                  const float* __restrict__ cls, const unsigned long long* __restrict__ key,
                  float* __restrict__ outrow, int n)
{
    const unsigned long long k = *key;
    const unsigned idx = 0xFFFFFFFFu - (unsigned)(k & 0xFFFFFFFFu);
    const float s = ord2f((unsigned)(k >> 32));
    const bool valid = (s >= SCORE_THR) && (idx < (unsigned)n);

    float4 b = make_float4(0.f, 0.f, 0.f, 0.f);
    if (valid) b = boxes[idx];

    if (blockIdx.x == 0 && threadIdx.x == 0) {
        if (valid) {
            outrow[0] = b.x; outrow[1] = b.y; outrow[2] = b.z; outrow[3] = b.w;
            outrow[4] = s;   outrow[5] = cls[idx];
        } else {
            outrow[0] = 0.f; outrow[1] = 0.f; outrow[2] = 0.f;
            outrow[3] = 0.f; outrow[4] = 0.f; outrow[5] = 0.f;
        }
    }
    if (!valid) return;

    const float barea = fmaxf(b.z - b.x, 0.f) * fmaxf(b.w - b.y, 0.f);
    for (int j = blockIdx.x * blockDim.x + threadIdx.x; j < n; j += gridDim.x * blockDim.x) {
        float4 bj = boxes[j];
        float iy1 = fmaxf(b.x, bj.x), ix1 = fmaxf(b.y, bj.y);
        float iy2 = fminf(b.z, bj.z), ix2 = fminf(b.w, bj.w);
        float inter = fmaxf(iy2 - iy1, 0.f) * fmaxf(ix2 - ix1, 0.f);
        float areaj = fmaxf(bj.z - bj.x, 0.f) * fmaxf(bj.w - bj.y, 0.f);
        float iou = inter / (barea + areaj - inter + 1e-9f);
        if (iou > IOU_THR || (unsigned)j == idx) sw[j] = -1.0f;
    }
}

// ================= host launcher =================
extern "C" void kernel_launch(void* const* d_in, const int* in_sizes, int n_in,
                              void* d_out, int out_size, void* d_ws, size_t ws_size,
                              hipStream_t stream)
{
    (void)in_sizes; (void)n_in; (void)out_size; (void)ws_size;
    const float* feats   = (const float*)d_in[0];
    const float* anchors = (const float*)d_in[1];
    float* out = (float*)d_out;

    char* ws = (char*)d_ws;
    unsigned long long* key = (unsigned long long*)ws;                      // 16-byte slot
    float*  sw    = (float*) (ws + 16);                                     // N floats
    float4* boxes = (float4*)(ws + 16 + sizeof(float)  * (size_t)N_CAND);   // N float4 (16B aligned)
    float*  cls   = (float*) (ws + 16 + sizeof(float)  * (size_t)N_CAND
                                      + sizeof(float4) * (size_t)N_CAND);   // N floats

    const int nBlocks = N_CAND / PER_BLOCK;   // 3610, exact
    yolo_decode<<<nBlocks, PER_BLOCK, 0, stream>>>(feats, anchors, sw, boxes, cls);

    for (int i = 0; i < MAXDET; ++i) {
        nms_reset   <<<1,   1,   0, stream>>>(key);
        nms_argmax  <<<512, 256, 0, stream>>>(sw, key, N_CAND);
        nms_suppress<<<904, 256, 0, stream>>>(sw, boxes, cls, key, out + i * 6, N_CAND);
    }
}